// NoisyTopKRouter_81844896792931
// MI455X (gfx1250) — compile-verified
//
#include <hip/hip_runtime.h>
#include <hip/hip_bf16.h>
#include <math.h>

typedef __attribute__((ext_vector_type(2))) float v2f;
typedef __attribute__((ext_vector_type(8))) float v8f;

#define D        2048
#define NE       8
#define KCHUNK   256
#define KSTRIDE  (KCHUNK + 4)       // padded column stride (floats) to spread LDS banks
#define WAVES    8
#define BLOCK    (WAVES * 32)
#define TPW      16                 // tokens per wave
#define TPB      (WAVES * TPW)      // tokens per block = 128

// ws layout: acc[0..7] = load, acc[8..15] = importance
__global__ void router_init_ws(float* acc) {
    if (threadIdx.x < 16) acc[threadIdx.x] = 0.0f;
}

__global__ __launch_bounds__(BLOCK) void router_main(
    const float* __restrict__ x, const float* __restrict__ Wg,
    float* __restrict__ out_dw, float* __restrict__ out_idx,
    float* __restrict__ acc, int nTok)
{
    __shared__ float Wlds[16 * KSTRIDE];     // ~16.3 KB: column-major W chunk, cols 8..15 = 0
    __shared__ float hTile[WAVES][16][16];   // 8 KB: per-wave 16x16 C spill
    __shared__ float blkAcc[16];             // block-local load/importance

    const int tid  = threadIdx.x;
    const int wave = tid >> 5;
    const int lane = tid & 31;
    const int col  = lane & 15;              // A row-in-tile == B column
    const int kb   = (lane >> 4) << 1;       // 0 or 2 (K sub-offset per half-wave)

    if (tid < 16) blkAcc[tid] = 0.0f;

    // Zero the padding columns 8..15 once (never rewritten).
    for (int i = tid; i < 8 * KSTRIDE; i += BLOCK) Wlds[8 * KSTRIDE + i] = 0.0f;

    const int tileBase = blockIdx.x * TPB + wave * TPW;
    const float* __restrict__ arow = x + (size_t)(tileBase + col) * D + kb;
    const int bBase = col * KSTRIDE + kb;    // per-lane B column base (floats)

    v8f c = {};
    for (int kc = 0; kc < D; kc += KCHUNK) {
        __syncthreads();
        // stage W[kc .. kc+KCHUNK) cols 0..7, transposed to column-major
        // (coalesced b32 global reads; padded stride avoids bank conflicts)
        for (int j = tid; j < KCHUNK * NE; j += BLOCK) {
            int k = j >> 3, e = j & 7;
            Wlds[e * KSTRIDE + k] = Wg[(size_t)kc * NE + j];
        }
        __syncthreads();

        #pragma unroll 8
        for (int kk = 0; kk < KCHUNK; kk += 4) {
            v2f a = *(const v2f*)(arow + kc + kk);        // global_load_b64
            v2f b = *(const v2f*)(Wlds + bBase + kk);     // ds_load_b64, cols 8..15 read zeros
            c = __builtin_amdgcn_wmma_f32_16x16x4_f32(
                    false, a, false, b, (short)0, c, false, false);
        }
    }

    // Spill C (h logits) to LDS: VGPR j, lanes 0-15 -> M=j, lanes 16-31 -> M=j+8
    #pragma unroll
    for (int j = 0; j < 8; ++j) {
        int row = j + ((lane >> 4) << 3);
        hTile[wave][row][col] = c[j];
    }
    __syncthreads();

    if (lane < TPW) {
        int token = tileBase + lane;
        if (token < nTok) {
            float hv[NE];
            #pragma unroll
            for (int e = 0; e < NE; ++e) hv[e] = hTile[wave][lane][e];

            // top-2, lowest-index tie-break (strict >)
            int i1 = 0; float v1 = hv[0];
            #pragma unroll
            for (int e = 1; e < NE; ++e) if (hv[e] > v1) { v1 = hv[e]; i1 = e; }
            int i2 = (i1 == 0) ? 1 : 0; float v2 = hv[i2];
            #pragma unroll
            for (int e = 0; e < NE; ++e)
                if (e != i1 && hv[e] > v2) { v2 = hv[e]; i2 = e; }

            // full softmax (importance contribution)
            float p[NE], s = 0.0f;
            #pragma unroll
            for (int e = 0; e < NE; ++e) { p[e] = __expf(hv[e] - v1); s += p[e]; }
            float inv = 1.0f / s;
            #pragma unroll
            for (int e = 0; e < NE; ++e) atomicAdd(&blkAcc[8 + e], p[e] * inv);
            atomicAdd(&blkAcc[i1], 1.0f);
            atomicAdd(&blkAcc[i2], 1.0f);

            // softmax over the two kept logits (v1 >= v2)
            float e2 = __expf(v2 - v1);
            float w1 = 1.0f / (1.0f + e2);
            float w2 = e2 * w1;
            out_dw [(size_t)token * 2 + 0] = w1;
            out_dw [(size_t)token * 2 + 1] = w2;
            out_idx[(size_t)token * 2 + 0] = (float)i1;
            out_idx[(size_t)token * 2 + 1] = (float)i2;
        }
    }
    __syncthreads();
    if (tid < 16) atomicAdd(&acc[tid], blkAcc[tid]);
}

__global__ void router_finalize(const float* __restrict__ acc, float* out_aux) {
    if (threadIdx.x == 0) {
        float lm = 0.0f, im = 0.0f;
        #pragma unroll
        for (int e = 0; e < NE; ++e) { lm += acc[e]; im += acc[8 + e]; }
        lm *= 0.125f; im *= 0.125f;
        float lv = 0.0f, iv = 0.0f;
        #pragma unroll
        for (int e = 0; e < NE; ++e) {
            float d0 = acc[e] - lm;      lv += d0 * d0;
            float d1 = acc[8 + e] - im;  iv += d1 * d1;
        }
        lv *= (1.0f / 7.0f);   // ddof = 1
        iv *= (1.0f / 7.0f);
        float cv = sqrtf(lv) / (lm + 1e-6f) + sqrtf(iv) / (im + 1e-6f);
        *out_aux = cv * 0.01f;
    }
}

extern "C" void kernel_launch(void* const* d_in, const int* in_sizes, int n_in,
                              void* d_out, int out_size, void* d_ws, size_t ws_size,
                              hipStream_t stream) {
    const float* x  = (const float*)d_in[0];
    const float* Wg = (const float*)d_in[1];
    // d_in[2] = W_noise (dead in eval mode), d_in[3] = top_k (== 2)

    const int nTok = in_sizes[0] / D;           // 32768
    float* out     = (float*)d_out;
    float* out_dw  = out;                        // [nTok, 2]
    float* out_idx = out + (size_t)nTok * 2;     // [nTok, 2] as float
    float* out_aux = out + (size_t)nTok * 4;     // scalar
    float* acc     = (float*)d_ws;               // 16 floats

    hipLaunchKernelGGL(router_init_ws, dim3(1), dim3(32), 0, stream, acc);

    const int nBlocks = (nTok + TPB - 1) / TPB;  // 256
    hipLaunchKernelGGL(router_main, dim3(nBlocks), dim3(BLOCK), 0, stream,
                       x, Wg, out_dw, out_idx, acc, nTok);

    hipLaunchKernelGGL(router_finalize, dim3(1), dim3(32), 0, stream, acc, out_aux);
}